// FluctuationExtractor_2413771621067
// MI455X (gfx1250) — compile-verified
//
#include <hip/hip_runtime.h>
#include <hip/hip_bf16.h>

// FluctuationExtractor, MI455X (gfx1250).
//
// Key insight: the per-timestep difference sums telescope, so z depends on at
// most 4 rows of X per batch. Total traffic ~1.4 MB -> pure launch-latency
// bound on a 23.3 TB/s part; one fused kernel, fp32 WMMA for the final GEMM
// (V_WMMA_F32_16X16X4_F32 keeps full fp32 precision, M=16 == batch).

#define B_   16
#define L_   4096
#define D_   512
#define OUT_ 512
#define ZPAD 4            // LDS row pad: stride 516 -> A-lane bank stride 4, 16 distinct banks

typedef __attribute__((ext_vector_type(2))) float v2f;
typedef __attribute__((ext_vector_type(8))) float v8f;

__global__ __launch_bounds__(256) void fluct_wmma_kernel(
    const float* __restrict__ X,        // (B, L, D)
    const int*   __restrict__ mask,     // (B, L) int32
    const float* __restrict__ alogits,  // (2,)
    const float* __restrict__ W,        // (OUT, D)
    const float* __restrict__ bias,     // (OUT,)
    float*       __restrict__ out)      // (B, OUT)
{
    __shared__ int   s_len[B_];
    __shared__ float s_z[B_][D_ + ZPAD];   // ~33 KB of the 320 KB WGP LDS

    const int tid = threadIdx.x;

    // ---- Phase 1: lengths[b] = sum_{j=1..L-1} mask[b][j] ----
    if (tid < B_) s_len[tid] = 0;
    __syncthreads();
    {
        const int b     = tid >> 4;          // 16 threads per batch row
        const int chunk = tid & 15;
        const int per   = (L_ - 1 + 15) / 16;
        const int j0    = 1 + chunk * per;
        const int j1    = (j0 + per < L_) ? (j0 + per) : L_;
        const int* mrow = mask + b * L_;
        int acc = 0;
        for (int j = j0; j < j1; ++j) acc += mrow[j];
        atomicAdd(&s_len[b], acc);
    }
    __syncthreads();

    // ---- alpha = softmax(alogits) (2 scalars; every thread computes) ----
    const float l0 = alogits[0], l1 = alogits[1];
    const float mx = fmaxf(l0, l1);
    const float e0 = __expf(l0 - mx), e1 = __expf(l1 - mx);
    const float inv = 1.0f / (e0 + e1);
    const float a1 = e0 * inv, a2 = e1 * inv;

    // ---- Phase 2: z into LDS via telescoped sums ----
    // S[j] = X[b, j+1, :]; n = s_len[b]
    //   n==0: z=0;  n==1: z=S[0]
    //   n>=2: z = (a1*(S[n-1]-S[0]) + a2*(S[n-2]+S[n-1]-S[0]-S[1])) / (n-1)
    for (int idx = tid; idx < B_ * D_; idx += 256) {
        const int b = idx >> 9;            // / D_
        const int d = idx & (D_ - 1);
        const int n = s_len[b];
        const float* Xb = X + (size_t)b * L_ * D_ + d;   // &X[b,0,d]
        float z;
        if (n == 0) {
            z = 0.0f;
        } else if (n == 1) {
            z = Xb[(size_t)1 * D_];                      // S[0]
        } else {
            const float S0  = Xb[(size_t)1 * D_];        // S[0]   = X[b,1,d]
            const float S1  = Xb[(size_t)2 * D_];        // S[1]   = X[b,2,d]
            const float Sa  = Xb[(size_t)(n - 1) * D_];  // S[n-2] = X[b,n-1,d]
            const float Sb  = Xb[(size_t)n * D_];        // S[n-1] = X[b,n,d]
            const float sum1 = Sb - S0;
            const float sum2 = (Sa + Sb) - (S0 + S1);
            z = (a1 * sum1 + a2 * sum2) / (float)(n - 1);
        }
        s_z[b][d] = z;
    }
    __syncthreads();

    // ---- Phase 3: out = z @ W^T + bias via V_WMMA_F32_16X16X4_F32 ----
    // One wave per 16-wide output tile: 4 blocks * 8 waves = 32 tiles = OUT/16.
    const int wave = tid >> 5;
    const int lane = tid & 31;
    const int n0   = (blockIdx.x * 8 + wave) * 16;

    const int half = lane >> 4;        // lane half selects K sub-pair
    const int ncol = lane & 15;        // N column (B/C/D) and M row (A)

    v8f c = {};
    const float* wrow = W + (size_t)(n0 + ncol) * D_;    // W[n0+ncol][*]
    #pragma unroll 4
    for (int k = 0; k < D_; k += 4) {
        const int kb = k + half * 2;
        // A (16x4 f32): VGPR0 = K {0|2}, VGPR1 = K {1|3} per lane half, row = ncol
        v2f a = *(const v2f*)&s_z[ncol][kb];
        // B (4x16 f32): lanes 0-15 hold K {0,1}, lanes 16-31 hold K {2,3};
        // B[k][n] = W[n0+n][k]
        v2f bb = *(const v2f*)(wrow + kb);
        c = __builtin_amdgcn_wmma_f32_16x16x4_f32(
                /*neg_a=*/false, a, /*neg_b=*/false, bb,
                /*c_mod=*/(short)0, c, /*reuse_a=*/false, /*reuse_b=*/false);
    }

    // D layout: lanes 0-15 -> M=v, N=ncol ; lanes 16-31 -> M=v+8, N=ncol
    const float bo = bias[n0 + ncol];
    #pragma unroll
    for (int v = 0; v < 8; ++v) {
        const int M = v + 8 * half;
        out[(size_t)M * OUT_ + n0 + ncol] = c[v] + bo;
    }
}

extern "C" void kernel_launch(void* const* d_in, const int* in_sizes, int n_in,
                              void* d_out, int out_size, void* d_ws, size_t ws_size,
                              hipStream_t stream) {
    const float* X       = (const float*)d_in[0];
    const int*   mask    = (const int*)  d_in[1];
    const float* alogits = (const float*)d_in[2];
    const float* W       = (const float*)d_in[3];
    const float* bias    = (const float*)d_in[4];
    float*       out     = (float*)d_out;

    fluct_wmma_kernel<<<dim3(4), dim3(256), 0, stream>>>(
        X, mask, alogits, W, bias, out);
}